// Model_29317446762539
// MI455X (gfx1250) — compile-verified
//
#include <hip/hip_runtime.h>
#include <hip/hip_bf16.h>

// ---------------------------------------------------------------------------
// GIN layer-0 + graph merge for MI455X (gfx1250, wave32, WMMA f32 16x16x4).
//
// Linearity reassociation: MLP/head matmuls are pushed *before* the edge and
// graph segment-sums, shrinking edge traffic from 801 -> 32 floats per edge.
// The concat([pos | seed=0 | feat]) @ W GEMM is split into two aligned,
// branch-free GEMMs (pos @ W[0:32], feat @ W[33:801]) - seed row drops out.
// ---------------------------------------------------------------------------

#define N_NODES 50000
#define D_POS   32
#define D_FEAT  768
#define D_IN    801      // 32 pos + 1 seed + 768 feat
#define D_HID   32
#define D_OUT   32
#define N_GRAPHS 128
#define M_TILES  (N_NODES / 16)          // 3125 (exact)
#define WAVES_PER_BLOCK 8

typedef __attribute__((ext_vector_type(2))) float v2f;
typedef __attribute__((ext_vector_type(8))) float v8f;

// ---------------------------------------------------------------------------
// Kernel 1: per-node  y = n_feat @ W1  and  z = n_feat @ Wp0   (both [N,32])
// One wave = one 16-node M tile; K swept in chunks of 4 (fp32 WMMA).
// acc is initialized to y (the "(1+eps)*h" self term of GIN).
// ---------------------------------------------------------------------------
__global__ __launch_bounds__(256) void input_gemm_kernel(
    const float* __restrict__ pos, const float* __restrict__ feat,
    const float* __restrict__ W1,  const float* __restrict__ Wp0,
    float* __restrict__ yBuf, float* __restrict__ accBuf, float* __restrict__ zBuf)
{
    const int wave = threadIdx.x >> 5;
    const int lane = threadIdx.x & 31;
    const int tile = blockIdx.x * WAVES_PER_BLOCK + wave;
    if (tile >= M_TILES) return;                       // wave-uniform exit

    const int  m0   = tile * 16;
    const int  col  = lane & 15;
    const bool hi   = lane >= 16;                      // this half covers K+2..K+3
    const int  node = m0 + col;                        // A-matrix row for this lane

    v8f cy0 = {}, cy1 = {}, cz0 = {}, cz1 = {};

    // --- Region 1: pos @ W[0:32, :]  (8 chunks, kb even -> aligned b64 A load)
    const float* posRow = pos + (size_t)node * D_POS;
    for (int k = 0; k < D_POS; k += 4) {
        const int kb = k + (hi ? 2 : 0);
        const v2f a = *(const v2f*)(posRow + kb);

        v2f by0, by1, bz0, bz1;
        by0.x = W1 [kb * 32 + col];        by0.y = W1 [(kb + 1) * 32 + col];
        by1.x = W1 [kb * 32 + 16 + col];   by1.y = W1 [(kb + 1) * 32 + 16 + col];
        bz0.x = Wp0[kb * 32 + col];        bz0.y = Wp0[(kb + 1) * 32 + col];
        bz1.x = Wp0[kb * 32 + 16 + col];   bz1.y = Wp0[(kb + 1) * 32 + 16 + col];

        cy0 = __builtin_amdgcn_wmma_f32_16x16x4_f32(false, a, false, by0, (short)0, cy0, false, false);
        cy1 = __builtin_amdgcn_wmma_f32_16x16x4_f32(false, a, false, by1, (short)0, cy1, false, false);
        cz0 = __builtin_amdgcn_wmma_f32_16x16x4_f32(false, a, false, bz0, (short)0, cz0, false, false);
        cz1 = __builtin_amdgcn_wmma_f32_16x16x4_f32(false, a, false, bz1, (short)0, cz1, false, false);
    }

    // --- Region 2: feat @ W[33:801, :]  (192 chunks, branch-free, aligned)
    const float* featRow = feat + (size_t)node * D_FEAT;
    for (int k = 0; k < D_FEAT; k += 4) {
        const int kb = k + (hi ? 2 : 0);
        const v2f a = *(const v2f*)(featRow + kb);

        const int r = (33 + kb) * 32;                  // W row offset (floats)
        v2f by0, by1, bz0, bz1;
        by0.x = W1 [r + col];        by0.y = W1 [r + 32 + col];
        by1.x = W1 [r + 16 + col];   by1.y = W1 [r + 48 + col];
        bz0.x = Wp0[r + col];        bz0.y = Wp0[r + 32 + col];
        bz1.x = Wp0[r + 16 + col];   bz1.y = Wp0[r + 48 + col];

        cy0 = __builtin_amdgcn_wmma_f32_16x16x4_f32(false, a, false, by0, (short)0, cy0, false, false);
        cy1 = __builtin_amdgcn_wmma_f32_16x16x4_f32(false, a, false, by1, (short)0, cy1, false, false);
        cz0 = __builtin_amdgcn_wmma_f32_16x16x4_f32(false, a, false, bz0, (short)0, cz0, false, false);
        cz1 = __builtin_amdgcn_wmma_f32_16x16x4_f32(false, a, false, bz1, (short)0, cz1, false, false);
    }

    // C/D layout: VGPR j -> row (hi?8:0)+j, col = lane&15 (tile1 adds +16)
    for (int j = 0; j < 8; ++j) {
        const int row = (hi ? 8 : 0) + j;
        const int off = (m0 + row) * 32 + col;
        yBuf[off]        = cy0[j];  accBuf[off]        = cy0[j];
        yBuf[off + 16]   = cy1[j];  accBuf[off + 16]   = cy1[j];
        zBuf[off]        = cz0[j];
        zBuf[off + 16]   = cz1[j];
    }
}

// ---------------------------------------------------------------------------
// Kernel 2: edge scatter  acc[dst] += y[src]   (32 floats/edge, f32 atomics)
// One thread = one (edge, 4-col group): 128B coalesced gather per edge.
// ---------------------------------------------------------------------------
__global__ __launch_bounds__(256) void edge_scatter_kernel(
    const float* __restrict__ yBuf, const int* __restrict__ src,
    const int* __restrict__ dst, float* __restrict__ accBuf, int nEdges)
{
    const int idx = blockIdx.x * blockDim.x + threadIdx.x;
    const int e = idx >> 3;
    if (e >= nEdges) return;
    const int c = (idx & 7) * 4;
    const int s = src[e], d = dst[e];
    const float4 v = *(const float4*)(yBuf + (size_t)s * 32 + c);
    float* p = accBuf + (size_t)d * 32 + c;
    atomicAdd(p + 0, v.x);
    atomicAdd(p + 1, v.y);
    atomicAdd(p + 2, v.z);
    atomicAdd(p + 3, v.w);
}

// ---------------------------------------------------------------------------
// Kernel 3: h  = relu(acc + b1)
//           h1 = relu(h @ W2 + b2)      (WMMA, K=32)
//           u  = h1 @ Wp1               (WMMA, K=32; transpose via LDS)
//           partial[gid] += u + z       (per-graph f32 atomics)
// ---------------------------------------------------------------------------
__global__ __launch_bounds__(256) void node_mlp_head_kernel(
    const float* __restrict__ accBuf, const float* __restrict__ zBuf,
    const float* __restrict__ W2, const float* __restrict__ bias1,
    const float* __restrict__ bias2, const float* __restrict__ Wp1,
    const int* __restrict__ gids, float* __restrict__ partial)
{
    __shared__ float hbuf[WAVES_PER_BLOCK][16 * 32];   // per-wave 16x32 h1 tile

    const int wave = threadIdx.x >> 5;
    const int lane = threadIdx.x & 31;
    const int tile = blockIdx.x * WAVES_PER_BLOCK + wave;
    if (tile >= M_TILES) return;                       // wave-uniform exit

    const int  m0   = tile * 16;
    const int  col  = lane & 15;
    const bool hi   = lane >= 16;
    const int  node = m0 + col;

    // ---- GEMM 1: relu(acc + b1) @ W2 ----
    const float* accRow = accBuf + (size_t)node * 32;
    v8f c10 = {}, c11 = {};
    for (int k = 0; k < 32; k += 4) {
        const int kb = k + (hi ? 2 : 0);
        const v2f av = *(const v2f*)(accRow + kb);
        const v2f bv = *(const v2f*)(bias1 + kb);
        v2f a;
        a.x = fmaxf(av.x + bv.x, 0.0f);
        a.y = fmaxf(av.y + bv.y, 0.0f);
        v2f b0, b1f;
        b0.x  = W2[kb * 32 + col];       b0.y  = W2[(kb + 1) * 32 + col];
        b1f.x = W2[kb * 32 + 16 + col];  b1f.y = W2[(kb + 1) * 32 + 16 + col];
        c10 = __builtin_amdgcn_wmma_f32_16x16x4_f32(false, a, false, b0,  (short)0, c10, false, false);
        c11 = __builtin_amdgcn_wmma_f32_16x16x4_f32(false, a, false, b1f, (short)0, c11, false, false);
    }

    // ---- h1 = relu(c1 + b2), stash in LDS to re-read in A layout ----
    const float bb0 = bias2[col];
    const float bb1 = bias2[16 + col];
    for (int j = 0; j < 8; ++j) {
        const int row = (hi ? 8 : 0) + j;
        hbuf[wave][row * 32 + col]      = fmaxf(c10[j] + bb0, 0.0f);
        hbuf[wave][row * 32 + 16 + col] = fmaxf(c11[j] + bb1, 0.0f);
    }
    // wave-local LDS RAW: DS ops are in-order per wave, compiler inserts waits

    // ---- GEMM 2: h1 @ Wp1 ----
    v8f u0 = {}, u1 = {};
    for (int k = 0; k < 32; k += 4) {
        const int kb = k + (hi ? 2 : 0);
        const v2f a = *(const v2f*)(&hbuf[wave][col * 32 + kb]);  // 8B aligned
        v2f b0, b1f;
        b0.x  = Wp1[kb * 32 + col];       b0.y  = Wp1[(kb + 1) * 32 + col];
        b1f.x = Wp1[kb * 32 + 16 + col];  b1f.y = Wp1[(kb + 1) * 32 + 16 + col];
        u0 = __builtin_amdgcn_wmma_f32_16x16x4_f32(false, a, false, b0,  (short)0, u0, false, false);
        u1 = __builtin_amdgcn_wmma_f32_16x16x4_f32(false, a, false, b1f, (short)0, u1, false, false);
    }

    // ---- per-graph accumulation of (u + z) ----
    for (int j = 0; j < 8; ++j) {
        const int row  = (hi ? 8 : 0) + j;
        const int nd   = m0 + row;
        const int g    = gids[nd];
        const int off  = nd * 32 + col;
        atomicAdd(&partial[g * 32 + col],      u0[j] + zBuf[off]);
        atomicAdd(&partial[g * 32 + 16 + col], u1[j] + zBuf[off + 16]);
    }
}

// ---------------------------------------------------------------------------
__global__ void zero_partial_kernel(float* __restrict__ partial) {
    const int i = blockIdx.x * blockDim.x + threadIdx.x;
    if (i < N_GRAPHS * D_OUT) partial[i] = 0.0f;
}

__global__ void finalize_kernel(const float* __restrict__ partial,
                                const float* __restrict__ bp0,
                                const float* __restrict__ bp1,
                                float* __restrict__ out) {
    const int i = blockIdx.x * blockDim.x + threadIdx.x;
    if (i < N_GRAPHS * D_OUT) {
        const int c = i & 31;
        out[i] = partial[i] + bp0[c] + bp1[c];
    }
}

// ---------------------------------------------------------------------------
extern "C" void kernel_launch(void* const* d_in, const int* in_sizes, int n_in,
                              void* d_out, int out_size, void* d_ws, size_t ws_size,
                              hipStream_t stream) {
    const float* pos  = (const float*)d_in[0];
    const float* feat = (const float*)d_in[1];
    const int*   src  = (const int*)  d_in[2];
    const int*   dst  = (const int*)  d_in[3];
    const int*   gids = (const int*)  d_in[4];
    const float* W1   = (const float*)d_in[5];
    const float* b1   = (const float*)d_in[6];
    const float* W2   = (const float*)d_in[7];
    const float* b2   = (const float*)d_in[8];
    const float* Wp0  = (const float*)d_in[9];
    const float* bp0  = (const float*)d_in[10];
    const float* Wp1  = (const float*)d_in[11];
    const float* bp1  = (const float*)d_in[12];
    float* out = (float*)d_out;
    const int nEdges = in_sizes[2];

    // Workspace layout (floats): y[N*32] | z[N*32] | acc[N*32] | partial[128*32]
    float* yBuf    = (float*)d_ws;
    float* zBuf    = yBuf + (size_t)N_NODES * 32;
    float* accBuf  = zBuf + (size_t)N_NODES * 32;
    float* partial = accBuf + (size_t)N_NODES * 32;

    const int gemmBlocks = (M_TILES + WAVES_PER_BLOCK - 1) / WAVES_PER_BLOCK; // 391

    zero_partial_kernel<<<(N_GRAPHS * D_OUT + 255) / 256, 256, 0, stream>>>(partial);
    input_gemm_kernel<<<gemmBlocks, 256, 0, stream>>>(pos, feat, W1, Wp0,
                                                      yBuf, accBuf, zBuf);
    edge_scatter_kernel<<<((size_t)nEdges * 8 + 255) / 256, 256, 0, stream>>>(
        yBuf, src, dst, accBuf, nEdges);
    node_mlp_head_kernel<<<gemmBlocks, 256, 0, stream>>>(accBuf, zBuf, W2, b1, b2,
                                                         Wp1, gids, partial);
    finalize_kernel<<<(N_GRAPHS * D_OUT + 255) / 256, 256, 0, stream>>>(
        partial, bp0, bp1, out);
}